// AttentionPooling_82257213653069
// MI455X (gfx1250) — compile-verified
//
#include <hip/hip_runtime.h>
#include <math.h>

typedef __attribute__((ext_vector_type(2))) float v2f;
typedef __attribute__((ext_vector_type(8))) float v8f;

#define DIM 128
#define RED_BLOCKS 256
#define RED_THREADS 256

// ---------------------------------------------------------------------------
// Kernel 1: scores[i] = dot(x[i,:], W) + b  via V_WMMA_F32_16X16X4_F32.
// One wave handles a 16-row tile. A = 16x4 slice of x (ISA A-layout:
// lane = M + 16*(K>=2); VGPR0 = even K, VGPR1 = odd K). B = W slice broadcast
// into all 16 columns, so every column of D equals the 16 scores.
// Lanes 0 and 16 (column N=0) write rows 0-7 / 8-15 of the tile.
// ---------------------------------------------------------------------------
__global__ __launch_bounds__(256) void score_wmma_kernel(
    const float* __restrict__ x, const float* __restrict__ W,
    const float* __restrict__ bias, float* __restrict__ scores, int n) {
  const int lane = threadIdx.x & 31;
  const int wave = threadIdx.x >> 5;
  const long tile = (long)blockIdx.x * 8 + wave;   // 16 rows per tile
  const long r0 = tile * 16;
  if (r0 >= n) return;                             // wave-uniform exit: EXEC stays all-1 for WMMA

  const int m  = lane & 15;
  const int kh = lane >> 4;                        // 0: K slots {0,1}, 1: K slots {2,3}
  long row = r0 + m;
  if (row >= n) row = n - 1;                       // clamp (duplicate load, write is guarded)
  const float* __restrict__ rowp = x + row * DIM;

  v8f acc = {};
#pragma unroll
  for (int kb = 0; kb < DIM; kb += 4) {
    const v2f a = *(const v2f*)(rowp + kb + 2 * kh);   // A[m][kb+2kh .. +1]
    const v2f bb = *(const v2f*)(W + kb + 2 * kh);     // B broadcast across columns
    acc = __builtin_amdgcn_wmma_f32_16x16x4_f32(
        /*neg_a=*/false, a, /*neg_b=*/false, bb,
        /*c_mod=*/(short)0, acc, /*reuse_a=*/false, /*reuse_b=*/false);
  }

  // D layout: VGPR v, lane l -> M = v + 8*(l>=16), N = l&15. Column 0 = lanes 0,16.
  if (m == 0) {
    const float b0 = bias[0];
#pragma unroll
    for (int v = 0; v < 8; ++v) {
      const long idx = r0 + v + 8 * kh;
      if (idx < n) scores[idx] = acc[v] + b0;
    }
  }
}

// ---------------------------------------------------------------------------
// Reduction kernels (scores are 8 MB -> L2 resident; these are cheap).
// ---------------------------------------------------------------------------
__global__ __launch_bounds__(RED_THREADS) void max_partial_kernel(
    const float* __restrict__ scores, int n, float* __restrict__ partial) {
  __shared__ float sm[RED_THREADS];
  float m = -INFINITY;
  for (int i = blockIdx.x * blockDim.x + threadIdx.x; i < n;
       i += gridDim.x * blockDim.x)
    m = fmaxf(m, scores[i]);
  sm[threadIdx.x] = m;
  __syncthreads();
  for (int s = RED_THREADS / 2; s > 0; s >>= 1) {
    if (threadIdx.x < s) sm[threadIdx.x] = fmaxf(sm[threadIdx.x], sm[threadIdx.x + s]);
    __syncthreads();
  }
  if (threadIdx.x == 0) partial[blockIdx.x] = sm[0];
}

__global__ __launch_bounds__(RED_THREADS) void max_final_kernel(
    const float* __restrict__ partial, float* __restrict__ gmax) {
  __shared__ float sm[RED_THREADS];
  sm[threadIdx.x] = partial[threadIdx.x];
  __syncthreads();
  for (int s = RED_THREADS / 2; s > 0; s >>= 1) {
    if (threadIdx.x < s) sm[threadIdx.x] = fmaxf(sm[threadIdx.x], sm[threadIdx.x + s]);
    __syncthreads();
  }
  if (threadIdx.x == 0) gmax[0] = sm[0];
}

// exp(s - gmax) written IN PLACE over scores; per-block partial sums.
__global__ __launch_bounds__(RED_THREADS) void expsum_partial_kernel(
    float* __restrict__ scores, int n, const float* __restrict__ gmaxp,
    float* __restrict__ partial) {
  __shared__ float sm[RED_THREADS];
  const float gmax = gmaxp[0];
  float s = 0.0f;
  for (int i = blockIdx.x * blockDim.x + threadIdx.x; i < n;
       i += gridDim.x * blockDim.x) {
    const float e = __expf(scores[i] - gmax);
    scores[i] = e;
    s += e;
  }
  sm[threadIdx.x] = s;
  __syncthreads();
  for (int st = RED_THREADS / 2; st > 0; st >>= 1) {
    if (threadIdx.x < st) sm[threadIdx.x] += sm[threadIdx.x + st];
    __syncthreads();
  }
  if (threadIdx.x == 0) partial[blockIdx.x] = sm[0];
}

__global__ __launch_bounds__(RED_THREADS) void sum_final_kernel(
    const float* __restrict__ partial, float* __restrict__ invZ) {
  __shared__ float sm[RED_THREADS];
  sm[threadIdx.x] = partial[threadIdx.x];
  __syncthreads();
  for (int s = RED_THREADS / 2; s > 0; s >>= 1) {
    if (threadIdx.x < s) sm[threadIdx.x] += sm[threadIdx.x + s];
    __syncthreads();
  }
  if (threadIdx.x == 0) invZ[0] = 1.0f / sm[0];
}

// ---------------------------------------------------------------------------
// Kernel 6: per-graph weighted pooling. batch is sorted, so block g binary
// searches its segment [start,end) and streams it: thread t accumulates
// sum_i escore[i] * x[i][t]; final write folds in 1/Z. No atomics, fully
// deterministic, every output element written (empty segments -> 0).
// ---------------------------------------------------------------------------
__global__ __launch_bounds__(DIM) void pool_kernel(
    const float* __restrict__ x, const int* __restrict__ batch,
    const float* __restrict__ escore, const float* __restrict__ invZp,
    float* __restrict__ out, int n) {
  const int g = blockIdx.x;
  // lower_bound(batch, g)
  int lo = 0, hi = n;
  while (lo < hi) {
    const int mid = (lo + hi) >> 1;
    if (batch[mid] < g) lo = mid + 1; else hi = mid;
  }
  const int start = lo;
  // lower_bound(batch, g+1)
  hi = n;
  while (lo < hi) {
    const int mid = (lo + hi) >> 1;
    if (batch[mid] < g + 1) lo = mid + 1; else hi = mid;
  }
  const int end = lo;

  const int t = threadIdx.x;
  float acc = 0.0f;
  for (int i = start; i < end; ++i) {
    long pf = (long)i + 8;
    if (pf >= n) pf = n - 1;
    __builtin_prefetch(x + pf * DIM + t, 0, 1);     // global_prefetch hint
    acc = fmaf(escore[i], x[(long)i * DIM + t], acc);
  }
  out[(long)g * DIM + t] = acc * invZp[0];
}

// ---------------------------------------------------------------------------
extern "C" void kernel_launch(void* const* d_in, const int* in_sizes, int n_in,
                              void* d_out, int out_size, void* d_ws, size_t ws_size,
                              hipStream_t stream) {
  const float* x     = (const float*)d_in[0];
  const int*   batch = (const int*)d_in[1];   // jnp.int64 downgrades to int32 (x64 off)
  const float* W     = (const float*)d_in[2];
  const float* bias  = (const float*)d_in[3];
  float*       out   = (float*)d_out;

  const int n          = in_sizes[0] / DIM;   // 2,000,000 nodes
  const int num_graphs = out_size / DIM;      // 16,384

  // Workspace layout (floats): [0,n) scores->exp values, then partials + scalars
  float* scores  = (float*)d_ws;
  float* partial = scores + n;
  float* gmax    = partial + RED_BLOCKS;
  float* invZ    = gmax + 1;

  // Pass 1: scores via WMMA (16 rows/wave, 8 waves/block)
  const int tiles  = (n + 15) / 16;
  const int blocks = (tiles + 7) / 8;
  score_wmma_kernel<<<blocks, 256, 0, stream>>>(x, W, bias, scores, n);

  // Softmax statistics (L2-resident)
  max_partial_kernel<<<RED_BLOCKS, RED_THREADS, 0, stream>>>(scores, n, partial);
  max_final_kernel<<<1, RED_THREADS, 0, stream>>>(partial, gmax);
  expsum_partial_kernel<<<RED_BLOCKS, RED_THREADS, 0, stream>>>(scores, n, gmax, partial);
  sum_final_kernel<<<1, RED_THREADS, 0, stream>>>(partial, invZ);

  // Pass 2: segment-pooled weighted sum, 1/Z folded into the write
  pool_kernel<<<num_graphs, DIM, 0, stream>>>(x, batch, scores, invZ, out, n);
}